// QuantizedBn2d_214748364822
// MI455X (gfx1250) — compile-verified
//
#include <hip/hip_runtime.h>
#include <hip/hip_bf16.h>

// Problem constants from the reference.
#define NUM_CLUSTERS 8
#define NUM_FEATURES 256
#define BATCH        32
#define HW           3136   // 56*56
#define NV           784    // HW / 4 float4 chunks per (b,c) tile
#define TPB          256

// Exact replica of the reference's integer pipeline.
// xi in [0,255]; dw, K fit in int32; subsum magnitude < 2^19, <<left (<=2) < 2^21.
// 64-bit product < 2^52; after >>31 result < 2^22 -> int32 for the rounding shift.
__device__ __forceinline__ float qbn_one(float xf, int dw, int K, int m0,
                                         int left, int right, int z3) {
    int xi = (int)xf;                       // x values are exact small ints
    int s  = xi * dw + K;                   // subsum
    long long a    = (long long)s * (long long)(1 << left);   // subsum << left
    long long prod = a * (long long)m0;
    long long nudge = (prod >= 0) ? (1LL << 30) : (1LL - (1LL << 30));
    int v = (int)((prod + nudge) >> 31);    // SaturatingRoundingDoublingHighMul
    int mask = (1 << right) - 1;            // RoundingDivideByPOT
    int rem  = v & mask;
    int thr  = (mask >> 1) + ((v < 0) ? 1 : 0);
    int r    = (v >> right) + ((rem > thr) ? 1 : 0);
    r += z3;
    r = (r < -128) ? -128 : (r > 127 ? 127 : r);
    return (float)r;
}

__global__ __launch_bounds__(TPB) void qbn2d_kernel(
    const float* __restrict__ x,   const int* __restrict__ bc,
    const int* __restrict__ weight, const int* __restrict__ bias,
    const int* __restrict__ z1p,   const int* __restrict__ z2p,
    const int* __restrict__ z3p,   const int* __restrict__ m0p,
    const int* __restrict__ shp,   float* __restrict__ out)
{
    __shared__ float4 buf[NV];      // 12544 B per block staging tile

    const int blk = blockIdx.x;             // [0, 32*256)
    const int b   = blk >> 8;
    const int c   = blk & (NUM_FEATURES - 1);

    // All of these are block-uniform -> SGPRs.
    const int cl  = bc[b];
    const int w   = weight[cl * NUM_FEATURES + c];
    const int bi  = bias[cl * NUM_FEATURES + c];
    const int z1  = z1p[cl];
    const int z2  = z2p[cl];
    const int z3  = z3p[cl];
    const int m0  = m0p[cl];
    const int sh  = shp[cl];
    const int left  = (sh < 0) ? -sh : 0;   // {0,1,2}
    const int right = (sh > 0) ?  sh : 0;   // [0,7]
    const int dw = w - z2;
    const int K  = z1 * z2 - w * z1 + bi;   // folded constant term

    const float4* __restrict__ xb = (const float4*)(x   + (size_t)blk * HW);
    float4*       __restrict__ ob = (float4*)      (out + (size_t)blk * HW);
    const int tid = threadIdx.x;

    // Low 32 bits of the generic pointer to LDS == LDS byte offset (aperture
    // layout: {shared_base_hi, lds_offset}) — what the async op's VDST expects.
    const unsigned lds_base = (unsigned)(uintptr_t)(&buf[0]);

    // Phase 1: issue async global->LDS B128 copies (ASYNCcnt-tracked TDM-lite
    // path; per-lane, EXEC-masked on the 784-not-multiple-of-256 tail).
    for (int j = tid; j < NV; j += TPB) {
        unsigned lds_addr = lds_base + (unsigned)(j * 16);
        unsigned goff     = (unsigned)(j * 16);
        asm volatile("global_load_async_to_lds_b128 %0, %1, %2"
                     :: "v"(lds_addr), "v"(goff), "s"(xb)
                     : "memory");
    }
    // Phase 2: drain this wave's async queue, then consume from LDS.
    asm volatile("s_wait_asynccnt 0" ::: "memory");

    for (int j = tid; j < NV; j += TPB) {
        float4 xv = buf[j];                 // ds_load_b128
        float4 ov;
        ov.x = qbn_one(xv.x, dw, K, m0, left, right, z3);
        ov.y = qbn_one(xv.y, dw, K, m0, left, right, z3);
        ov.z = qbn_one(xv.z, dw, K, m0, left, right, z3);
        ov.w = qbn_one(xv.w, dw, K, m0, left, right, z3);
        ob[j] = ov;                         // global_store_b128
    }
}

extern "C" void kernel_launch(void* const* d_in, const int* in_sizes, int n_in,
                              void* d_out, int out_size, void* d_ws, size_t ws_size,
                              hipStream_t stream) {
    const float* x      = (const float*)d_in[0];
    const int*   bc     = (const int*)  d_in[1];
    const int*   weight = (const int*)  d_in[2];
    const int*   bias   = (const int*)  d_in[3];
    const int*   z1     = (const int*)  d_in[4];
    const int*   z2     = (const int*)  d_in[5];
    const int*   z3     = (const int*)  d_in[6];
    const int*   M0     = (const int*)  d_in[7];
    const int*   shift  = (const int*)  d_in[8];
    float* out = (float*)d_out;

    dim3 grid(BATCH * NUM_FEATURES);   // 8192 blocks, one (b,c) tile each
    qbn2d_kernel<<<grid, TPB, 0, stream>>>(x, bc, weight, bias,
                                           z1, z2, z3, M0, shift, out);
}